// MGRANet_student_89730456748872
// MI455X (gfx1250) — compile-verified
//
#include <hip/hip_runtime.h>
#include <hip/hip_bf16.h>

// Problem constants (match reference)
static constexpr int B  = 4;
static constexpr int C  = 256;
static constexpr int H  = 128;
static constexpr int W  = 128;
static constexpr int HW = H * W;       // 16384
static constexpr int P  = 2048;
static constexpr float THR = 0.8f;

typedef float v2f __attribute__((ext_vector_type(2)));
typedef float v8f __attribute__((ext_vector_type(8)));
typedef unsigned int u32x4 __attribute__((ext_vector_type(4)));
typedef int i32x4 __attribute__((ext_vector_type(4)));
typedef int i32x8 __attribute__((ext_vector_type(8)));

#if defined(__has_builtin)
#if __has_builtin(__builtin_amdgcn_tensor_load_to_lds) && __has_builtin(__builtin_amdgcn_s_wait_tensorcnt)
#define USE_TDM 1
#endif
#endif
#ifndef USE_TDM
#define USE_TDM 0
#endif

// ---------------------------------------------------------------------------
// TDM: issue a 2D tile DMA global->LDS.
// Tile: tileRows rows of tileCols fp32, row stride rowStrideElems (fp32 units).
// LDS layout produced: row pitch = tileCols + 4 DWORDs via the descriptor's
// pad feature (pad_interval code 5 = 64 DWORDs, pad_amount code 3 = 4 DWORDs),
// matching the __shared__ float T[rows][68] tiles used by the compute loops.
// D# bit packing per CDNA5 ISA ch.8 (group0 128b, group1 256b).
// ---------------------------------------------------------------------------
__device__ __forceinline__ void tdm_load_tile(const void* gptr, unsigned ldsOff,
                                              unsigned tileCols, unsigned tileRows,
                                              unsigned rowStrideElems,
                                              unsigned tensorCols, unsigned tensorRows) {
#if USE_TDM
    unsigned long long ga = (unsigned long long)(uintptr_t)gptr;
    u32x4 g0;
    g0[0] = 1u;                                   // count=1 (valid), user mode
    g0[1] = ldsOff;                               // lds_addr (bytes)
    g0[2] = (unsigned)(ga & 0xFFFFFFFFu);         // global_addr[31:0]
    g0[3] = (unsigned)((ga >> 32) & 0x01FFFFFFu)  // global_addr[56:32]
            | (2u << 30);                         // type = 2 ("image")
    i32x8 g1;
    g1[0] = (int)((2u << 16)      // data_size = 2 -> 4 bytes
                | (1u << 20)      // pad_enable
                | (5u << 22)      // pad_interval: 2^(5+1) = 64 DWORDs
                | (3u << 25));    // pad_amount: 3+1 = 4 DWORDs
    g1[1] = (int)((tensorCols & 0xFFFFu) << 16);                        // tensor_dim0[15:0]
    g1[2] = (int)((tensorCols >> 16) | ((tensorRows & 0xFFFFu) << 16)); // dim0 hi | dim1 lo
    g1[3] = (int)((tensorRows >> 16) | (tileCols << 16));               // dim1 hi | tile_dim0
    g1[4] = (int)tileRows;                                              // tile_dim1 (tile_dim2=0)
    g1[5] = (int)rowStrideElems;                                        // tensor_dim0_stride[31:0]
    g1[6] = 0;
    g1[7] = 0;
    i32x4 gz4 = {0, 0, 0, 0};
#if defined(__clang_major__) && (__clang_major__ >= 23)
    i32x8 gz8 = {0, 0, 0, 0, 0, 0, 0, 0};
    __builtin_amdgcn_tensor_load_to_lds(g0, g1, gz4, gz4, gz8, 0);
#else
    __builtin_amdgcn_tensor_load_to_lds(g0, g1, gz4, gz4, 0);
#endif
#else
    (void)gptr; (void)ldsOff; (void)tileCols; (void)tileRows;
    (void)rowStrideElems; (void)tensorCols; (void)tensorRows;
#endif
}

__device__ __forceinline__ void tdm_wait_all() {
#if USE_TDM
    __builtin_amdgcn_s_wait_tensorcnt(0);
#endif
}

__device__ __forceinline__ unsigned lds_offset_of(const void* p) {
    // Generic pointer to LDS: low 32 bits are the LDS byte offset (aperture rule).
    return (unsigned)(uintptr_t)p;
}

// ---------------------------------------------------------------------------
// Kernel 1: per-batch top-k via full bitonic sort in LDS.
// 16384 keys * 8B = 128 KB dynamic LDS (CDNA5 WGP has 320 KB).
// key = (float_bits(value) << 32) | (0xFFFFFFFF - index): descending sort =>
// largest value first, ties -> smaller index first (jax.lax.top_k rule).
// ---------------------------------------------------------------------------
__global__ __launch_bounds__(1024)
void topk_kernel(const float* __restrict__ edge, int* __restrict__ idx) {
    extern __shared__ unsigned long long keys[];   // [HW]
    const int b   = blockIdx.x;
    const int tid = threadIdx.x;
    const int nt  = blockDim.x;
    const float* e = edge + (size_t)b * HW;

    for (int i = tid; i < HW; i += nt) {
        float v = e[i];
        if (v < THR) v = 0.0f;
        unsigned vb = __float_as_uint(v);
        keys[i] = ((unsigned long long)vb << 32) |
                  (unsigned long long)(0xFFFFFFFFu - (unsigned)i);
    }
    __syncthreads();

    for (int k = 2; k <= HW; k <<= 1) {
        for (int j = k >> 1; j > 0; j >>= 1) {
            for (int t = tid; t < HW / 2; t += nt) {
                int i  = ((t & ~(j - 1)) << 1) | (t & (j - 1));
                int p  = i | j;
                unsigned long long a = keys[i];
                unsigned long long c = keys[p];
                bool descBlock = ((i & k) == 0);
                if (descBlock ? (a < c) : (a > c)) {
                    keys[i] = c;
                    keys[p] = a;
                }
            }
            __syncthreads();
        }
    }

    for (int t = tid; t < P; t += nt) {
        idx[b * P + t] = (int)(0xFFFFFFFFu - (unsigned)(keys[t] & 0xFFFFFFFFu));
    }
}

// ---------------------------------------------------------------------------
// Kernel 2: gather, transposed layout: gT[b][c][p] = x[b][c][idx[b][p]].
// Writes fully coalesced along p; gathered x reads absorbed by the 192MB L2
// (whole x is 64MB). The [C][P] layout makes GEMM1's B tile contiguous-in-K,
// so WMMA B fragments become single ds_load_b64s.
// ---------------------------------------------------------------------------
__global__ __launch_bounds__(256)
void gather_kernel(const float* __restrict__ x, const int* __restrict__ idx,
                   float* __restrict__ gT) {
    const int c = blockIdx.x;
    const int b = blockIdx.y;
    const float* xc = x + ((size_t)b * C + c) * HW;
    float*       gc = gT + ((size_t)b * C + c) * P;
    for (int p = threadIdx.x; p < P; p += 256) {
        gc[p] = xc[idx[b * P + p]];
    }
}

// ---------------------------------------------------------------------------
// WMMA GEMM: block = 256 threads = 8 waves; block tile 128(M) x 64(N), K
// staged in chunks of 64 by the Tensor Data Mover, double-buffered so the
// DMA for stage s+1 runs under stage s's WMMA block. Each wave computes a
// 16x64 strip = 4 v8f accumulators via V_WMMA_F32_16X16X4_F32.
// Both A and B tiles land K-contiguous ([m][k] / [n][k], pitch 68), so every
// fragment is one contiguous 8-byte LDS load; 68 mod 64 banks = 4 keeps the
// 16-lane column reads conflict-free.
// ---------------------------------------------------------------------------
static constexpr int TM = 128;
static constexpr int TN = 64;
static constexpr int TK = 64;
static constexpr int LDP = TK + 4;   // row pitch 68 = TDM pad layout

// GEMM1: z[b][o][c] = relu(BN_o(sum_p A[o][p] * gT[b][c][p])) + gT[b][c][o]
// A: conv_adj_w [P][P]. M=o (P), N=c (C), K=p (P). z stored [P][C].
__global__ __launch_bounds__(256)
void gemm_adj_kernel(const float* __restrict__ A,
                     const float* __restrict__ gT,
                     const float* __restrict__ gamma,
                     const float* __restrict__ beta,
                     float* __restrict__ z) {
    __shared__ float As[2][TM][LDP];      // A tile [m][k]
    __shared__ float Bt[2][TN][LDP];      // B tile [n][k]

    const int b      = blockIdx.z;
    const int mBlock = blockIdx.y * TM;   // over P (node o)
    const int nBlock = blockIdx.x * TN;   // over C (channel)
    const float* G = gT + (size_t)b * C * P;
    float*       Z = z + (size_t)b * P * C;

    const int tid  = threadIdx.x;
    const int lane = tid & 31;
    const int wave = tid >> 5;
    const int laneN = lane & 15;
    const int kSel  = (lane >> 4) * 2;    // frag: lanes 0-15 hold K=0,1; 16-31 K=2,3

    const unsigned asOff[2] = { lds_offset_of(&As[0][0][0]), lds_offset_of(&As[1][0][0]) };
    const unsigned btOff[2] = { lds_offset_of(&Bt[0][0][0]), lds_offset_of(&Bt[1][0][0]) };

    v8f acc[TN / 16] = {};
    const int nStages = P / TK;
    int cur = 0;

#if USE_TDM
    if (wave == 0) {
        tdm_load_tile(A + (size_t)mBlock * P, asOff[0], TK, TM, P, P, TM);
        tdm_load_tile(G + (size_t)nBlock * P, btOff[0], TK, TN, P, P, TN);
    }
#endif
    for (int s = 0; s < nStages; ++s) {
#if USE_TDM
        if (wave == 0) tdm_wait_all();     // buf[cur] ready
        __syncthreads();                   // publish to all waves
        if (wave == 0 && s + 1 < nStages) {
            int k0 = (s + 1) * TK;
            tdm_load_tile(A + (size_t)mBlock * P + k0, asOff[cur ^ 1], TK, TM, P, P, TM);
            tdm_load_tile(G + (size_t)nBlock * P + k0, btOff[cur ^ 1], TK, TN, P, P, TN);
        }
#else
        {
            int k0 = s * TK;
            for (int i = tid; i < TM * TK; i += 256) {
                int r = i >> 6, kc = i & 63;
                As[cur][r][kc] = A[(size_t)(mBlock + r) * P + k0 + kc];
            }
            for (int i = tid; i < TN * TK; i += 256) {
                int n = i >> 6, kc = i & 63;
                Bt[cur][n][kc] = G[(size_t)(nBlock + n) * P + k0 + kc];
            }
        }
        __syncthreads();
#endif
        const int mRow = wave * 16 + laneN;
#pragma unroll 4
        for (int ks = 0; ks < TK; ks += 4) {
            v2f a;
            a.x = As[cur][mRow][ks + kSel];
            a.y = As[cur][mRow][ks + kSel + 1];
#pragma unroll
            for (int nt = 0; nt < TN / 16; ++nt) {
                v2f bf;
                bf.x = Bt[cur][nt * 16 + laneN][ks + kSel];
                bf.y = Bt[cur][nt * 16 + laneN][ks + kSel + 1];
                acc[nt] = __builtin_amdgcn_wmma_f32_16x16x4_f32(
                    false, a, false, bf, (short)0, acc[nt], false, false);
            }
        }
        __syncthreads();                   // done reading buf[cur]
        cur ^= 1;
    }

    // Epilogue: BN over M (node) axis, ReLU, residual add of gT[c][o].
    const float inv = rsqrtf(1.0f + 1e-5f);
    const int mWave = mBlock + wave * 16;
    const int mHalf = (lane >> 4) * 8;     // C/D layout: VGPR r -> rows r, r+8
#pragma unroll
    for (int nt = 0; nt < TN / 16; ++nt) {
#pragma unroll
        for (int r = 0; r < 8; ++r) {
            int m = mWave + mHalf + r;     // node o
            int n = nBlock + nt * 16 + laneN;  // channel c
            float v = acc[nt][r] * inv * gamma[m] + beta[m];
            v = v > 0.0f ? v : 0.0f;
            Z[(size_t)m * C + n] = v + G[(size_t)n * P + m];
        }
    }
}

// GEMM2 + scatter: out[b][d][idx[b][p]] = relu(BN_d(sum_c Wg[d][c]*z[b][p][c]))
// M=d (C), N=p (P), K=c (C). z is [P][C] so the B tile is K-contiguous too.
__global__ __launch_bounds__(256)
void gemm_wg_scatter_kernel(const float* __restrict__ Wg,
                            const float* __restrict__ z,
                            const float* __restrict__ gamma,
                            const float* __restrict__ beta,
                            const int* __restrict__ idx,
                            float* __restrict__ out) {
    __shared__ float As[2][TM][LDP];      // Wg tile [m][k]
    __shared__ float Bt[2][TN][LDP];      // Z tile  [n][k]

    const int b      = blockIdx.z;
    const int mBlock = blockIdx.y * TM;   // over C (channel d)
    const int nBlock = blockIdx.x * TN;   // over P (node p)
    const float* Z = z + (size_t)b * P * C;

    const int tid  = threadIdx.x;
    const int lane = tid & 31;
    const int wave = tid >> 5;
    const int laneN = lane & 15;
    const int kSel  = (lane >> 4) * 2;

    const unsigned asOff[2] = { lds_offset_of(&As[0][0][0]), lds_offset_of(&As[1][0][0]) };
    const unsigned btOff[2] = { lds_offset_of(&Bt[0][0][0]), lds_offset_of(&Bt[1][0][0]) };

    v8f acc[TN / 16] = {};
    const int nStages = C / TK;
    int cur = 0;

#if USE_TDM
    if (wave == 0) {
        tdm_load_tile(Wg + (size_t)mBlock * C, asOff[0], TK, TM, C, C, TM);
        tdm_load_tile(Z + (size_t)nBlock * C, btOff[0], TK, TN, C, C, TN);
    }
#endif
    for (int s = 0; s < nStages; ++s) {
#if USE_TDM
        if (wave == 0) tdm_wait_all();
        __syncthreads();
        if (wave == 0 && s + 1 < nStages) {
            int k0 = (s + 1) * TK;
            tdm_load_tile(Wg + (size_t)mBlock * C + k0, asOff[cur ^ 1], TK, TM, C, C, TM);
            tdm_load_tile(Z + (size_t)nBlock * C + k0, btOff[cur ^ 1], TK, TN, C, C, TN);
        }
#else
        {
            int k0 = s * TK;
            for (int i = tid; i < TM * TK; i += 256) {
                int r = i >> 6, kc = i & 63;
                As[cur][r][kc] = Wg[(size_t)(mBlock + r) * C + k0 + kc];
            }
            for (int i = tid; i < TN * TK; i += 256) {
                int n = i >> 6, kc = i & 63;
                Bt[cur][n][kc] = Z[(size_t)(nBlock + n) * C + k0 + kc];
            }
        }
        __syncthreads();
#endif
        const int mRow = wave * 16 + laneN;
#pragma unroll 4
        for (int ks = 0; ks < TK; ks += 4) {
            v2f a;
            a.x = As[cur][mRow][ks + kSel];
            a.y = As[cur][mRow][ks + kSel + 1];
#pragma unroll
            for (int nt = 0; nt < TN / 16; ++nt) {
                v2f bf;
                bf.x = Bt[cur][nt * 16 + laneN][ks + kSel];
                bf.y = Bt[cur][nt * 16 + laneN][ks + kSel + 1];
                acc[nt] = __builtin_amdgcn_wmma_f32_16x16x4_f32(
                    false, a, false, bf, (short)0, acc[nt], false, false);
            }
        }
        __syncthreads();
        cur ^= 1;
    }

    // Epilogue: BN over M (channel) axis, ReLU, scatter into out at topk cols.
    const float inv = rsqrtf(1.0f + 1e-5f);
    const int mWave = mBlock + wave * 16;
    const int mHalf = (lane >> 4) * 8;
#pragma unroll
    for (int nt = 0; nt < TN / 16; ++nt) {
#pragma unroll
        for (int r = 0; r < 8; ++r) {
            int m = mWave + mHalf + r;              // channel d
            int n = nBlock + nt * 16 + laneN;       // node p
            int col = idx[b * P + n];
            float v = acc[nt][r] * inv * gamma[m] + beta[m];
            v = v > 0.0f ? v : 0.0f;
            out[((size_t)b * C + m) * HW + col] = v;
        }
    }
}

// ---------------------------------------------------------------------------
// Launch.  inputs: 0:x 1:edge 2:conv_adj_w 3:bn_adj_gamma 4:bn_adj_beta
//                  5:conv_wg_w 6:bn_wg_gamma 7:bn_wg_beta
// ---------------------------------------------------------------------------
extern "C" void kernel_launch(void* const* d_in, const int* in_sizes, int n_in,
                              void* d_out, int out_size, void* d_ws, size_t ws_size,
                              hipStream_t stream) {
    (void)in_sizes; (void)n_in; (void)out_size; (void)ws_size;

    const float* x     = (const float*)d_in[0];
    const float* edge  = (const float*)d_in[1];
    const float* adj_w = (const float*)d_in[2];
    const float* adj_g = (const float*)d_in[3];
    const float* adj_b = (const float*)d_in[4];
    const float* wg_w  = (const float*)d_in[5];
    const float* wg_g  = (const float*)d_in[6];
    const float* wg_b  = (const float*)d_in[7];
    float*       out   = (float*)d_out;

    char* ws = (char*)d_ws;
    int*   idx = (int*)ws;                                  // B*P ints   (32 KB)
    float* gT  = (float*)(ws + 32768);                      // B*C*P f32  (8 MB)
    float* z   = gT + (size_t)B * C * P;                    // B*P*C f32  (8 MB)

    // 1) passthrough copy of x into out (scatter overwrites P columns later)
    hipMemcpyAsync(out, x, (size_t)B * C * HW * sizeof(float),
                   hipMemcpyDeviceToDevice, stream);

    // 2) top-k indices per batch (bitonic sort of 16384 keys in 128KB LDS)
    topk_kernel<<<B, 1024, HW * sizeof(unsigned long long), stream>>>(edge, idx);

    // 3) gather sampled features, transposed: gT[B][C][P]
    gather_kernel<<<dim3(C, B), 256, 0, stream>>>(x, idx, gT);

    // 4) node-mixing GEMM + BN + ReLU + residual  ->  z[B][P][C]
    gemm_adj_kernel<<<dim3(C / TN, P / TM, B), 256, 0, stream>>>(
        adj_w, gT, adj_g, adj_b, z);

    // 5) feature-mixing GEMM + BN + ReLU + fused scatter into out
    gemm_wg_scatter_kernel<<<dim3(P / TN, C / TM, B), 256, 0, stream>>>(
        wg_w, z, wg_g, wg_b, idx, out);
}